// ScaledDotProductAttention_71133248356959
// MI455X (gfx1250) — compile-verified
//
#include <hip/hip_runtime.h>

typedef __attribute__((ext_vector_type(2))) float v2f;
typedef __attribute__((ext_vector_type(4))) float v4f;
typedef __attribute__((ext_vector_type(8))) float v8f;
typedef __attribute__((ext_vector_type(4))) int   v4i;

#define BATCH   16
#define LSEQ    2048
#define DHEAD   64
#define QT      16            // q rows per workgroup
#define SSTRIDE 2052          // LDS row stride in floats (16B-aligned rows, conflict-free halves)
#define NWAVE   8

__global__ __launch_bounds__(256) void sdpa_wmma_f32_kernel(
    const float* __restrict__ qp, const float* __restrict__ kp,
    const float* __restrict__ vp, const int* __restrict__ diagp,
    const unsigned char* __restrict__ maskp,
    float* __restrict__ outp, float* __restrict__ attnp)
{
  __shared__ float s_buf[QT * SSTRIDE];   // scores -> exp -> P
  __shared__ float s_part[NWAVE * 256];   // split-K partial O tiles
  __shared__ float s_max[QT];
  __shared__ float s_sum[QT];

  const int tid  = threadIdx.x;
  const int wave = tid >> 5;
  const int lane = tid & 31;
  const int m    = lane & 15;   // row (A) / col (B,C) within 16
  const int kh   = lane >> 4;   // K-half select for f32 wmma fragments

  const int b  = blockIdx.x >> 7;          // / (LSEQ/QT) == 128
  const int q0 = (blockIdx.x & 127) << 4;

  const size_t bLL   = (size_t)b * LSEQ * LSEQ;
  const float* qbase = qp + ((size_t)b * LSEQ + q0) * DHEAD;
  const float* kbase = kp + (size_t)b * LSEQ * DHEAD;
  const float* vbase = vp + (size_t)b * LSEQ * DHEAD;

  // ---------------- Phase 1: S = QK^T / T into LDS (no masks here) ----------------
  // A (16x4 f32) layout: lanes 0-15 hold row m, K={0,1}; lanes 16-31 K={2,3}.
  v2f afrag[16];
  {
    const float* qrow = qbase + (size_t)m * DHEAD + 2 * kh;
    #pragma unroll
    for (int kk = 0; kk < 16; ++kk)
      afrag[kk] = *(const v2f*)(qrow + kk * 4);
  }

  for (int it = 0; it < 16; ++it) {
    const int kt  = wave + it * NWAVE;     // k tile 0..127
    const int kc0 = kt << 4;
    // B (4x16 f32) = K^T tile: lane holds col m, K rows {2kh, 2kh+1}
    const float* krow = kbase + (size_t)(kc0 + m) * DHEAD + 2 * kh;
    v8f acc = {0.f,0.f,0.f,0.f,0.f,0.f,0.f,0.f};
    #pragma unroll
    for (int kk = 0; kk < 16; ++kk) {
      v2f bfrag = *(const v2f*)(krow + kk * 4);
      acc = __builtin_amdgcn_wmma_f32_16x16x4_f32(
          false, afrag[kk], false, bfrag, (short)0, acc, false, false);
    }
    // C/D layout: VGPR v -> row v + 8*kh, col m
    #pragma unroll
    for (int v = 0; v < 8; ++v)
      s_buf[(v + 8 * kh) * SSTRIDE + kc0 + m] = acc[v] * 0.125f;  // 1/TEMPERATURE
  }
  __syncthreads();

  // ---------------- Phase 1.5: streaming mask pass (vectorized, NT) ----------------
  // Consecutive-column ownership: diag as b128, mask as b32, LDS as v4f.
  for (int idx = tid; idx < QT * (LSEQ / 4); idx += 256) {
    const int r  = idx >> 9;               // / 512
    const int c4 = (idx & 511) << 2;
    const size_t g = bLL + (size_t)(q0 + r) * LSEQ + c4;
    v4i dg = __builtin_nontemporal_load((const v4i*)(diagp + g));
    unsigned mk = __builtin_nontemporal_load((const unsigned*)(maskp + g));
    float* sp = s_buf + r * SSTRIDE + c4;
    v4f s = *(v4f*)sp;
    if (mk & 0x000000FFu) s.x = -__builtin_inff();
    if (mk & 0x0000FF00u) s.y = -__builtin_inff();
    if (mk & 0x00FF0000u) s.z = -__builtin_inff();
    if (mk & 0xFF000000u) s.w = -__builtin_inff();
    if (dg.x == 0) s.x = -1e32f;
    if (dg.y == 0) s.y = -1e32f;
    if (dg.z == 0) s.z = -1e32f;
    if (dg.w == 0) s.w = -1e32f;
    *(v4f*)sp = s;
  }
  __syncthreads();

  // ---------------- Phase 2: row max, then exp in-place + sum ----------------
  for (int r = wave; r < QT; r += NWAVE) {
    v4f* row = (v4f*)(s_buf + r * SSTRIDE);
    float mx = -__builtin_inff();
    for (int i = lane; i < LSEQ / 4; i += 32) {
      v4f t = row[i];
      mx = fmaxf(mx, fmaxf(fmaxf(t.x, t.y), fmaxf(t.z, t.w)));
    }
    #pragma unroll
    for (int off = 16; off >= 1; off >>= 1)
      mx = fmaxf(mx, __shfl_xor(mx, off, 32));
    float sm = 0.f;
    for (int i = lane; i < LSEQ / 4; i += 32) {
      v4f t = row[i];
      t.x = __expf(t.x - mx); t.y = __expf(t.y - mx);
      t.z = __expf(t.z - mx); t.w = __expf(t.w - mx);
      row[i] = t;                                   // keep exp(s-m) in LDS
      sm += (t.x + t.y) + (t.z + t.w);
    }
    #pragma unroll
    for (int off = 16; off >= 1; off >>= 1)
      sm += __shfl_xor(sm, off, 32);
    if (lane == 0) { s_max[r] = mx; s_sum[r] = sm; }
  }
  __syncthreads();

  // ---------------- Phase 3: normalize P in LDS; stream attn out (b128 NT) ----------------
  for (int idx = tid; idx < QT * (LSEQ / 4); idx += 256) {
    const int r  = idx >> 9;
    const int c4 = (idx & 511) << 2;
    const float inv = 1.0f / s_sum[r];
    float* sp = s_buf + r * SSTRIDE + c4;
    v4f p = *(v4f*)sp;
    p.x *= inv; p.y *= inv; p.z *= inv; p.w *= inv;
    *(v4f*)sp = p;
    __builtin_nontemporal_store(p, (v4f*)(attnp + bLL + (size_t)(q0 + r) * LSEQ + c4));
  }
  __syncthreads();

  // ---------------- Phase 4: O = P V (split-K x2, 4 col tiles) ----------------
  {
    const int tile = wave & 3;       // output col tile (16 of 64 cols)
    const int half = wave >> 2;      // K half
    const int d0   = tile << 4;
    v8f acc = {0.f,0.f,0.f,0.f,0.f,0.f,0.f,0.f};
    const float* vcol = vbase + d0 + m;
    const float* prow = s_buf + m * SSTRIDE + 2 * kh;
    const int k_lo = half * (LSEQ / 2), k_hi = k_lo + (LSEQ / 2);
    for (int k4 = k_lo; k4 < k_hi; k4 += 4) {
      v2f a = *(const v2f*)(prow + k4);           // A = P tile from LDS
      v2f bfrag;                                  // B = V tile (rows 2kh,2kh+1)
      bfrag.x = vcol[(size_t)(k4 + 2 * kh)     * DHEAD];
      bfrag.y = vcol[(size_t)(k4 + 2 * kh + 1) * DHEAD];
      acc = __builtin_amdgcn_wmma_f32_16x16x4_f32(
          false, a, false, bfrag, (short)0, acc, false, false);
    }
    float* part = s_part + wave * 256 + lane * 8;
    #pragma unroll
    for (int v = 0; v < 8; ++v) part[v] = acc[v];
  }
  __syncthreads();

  if (wave < 4) {
    const int tile = wave;
    const float* p0 = s_part + tile * 256 + lane * 8;
    const float* p1 = s_part + (4 + tile) * 256 + lane * 8;
    #pragma unroll
    for (int v = 0; v < 8; ++v) {
      const int row = v + 8 * kh;
      float o = p0[v] + p1[v];
      outp[((size_t)b * LSEQ + q0 + row) * DHEAD + (tile << 4) + m] = o;
    }
  }
}

extern "C" void kernel_launch(void* const* d_in, const int* in_sizes, int n_in,
                              void* d_out, int out_size, void* d_ws, size_t ws_size,
                              hipStream_t stream) {
  const float*         q    = (const float*)d_in[0];
  const float*         k    = (const float*)d_in[1];
  const float*         v    = (const float*)d_in[2];
  const int*           diag = (const int*)d_in[3];
  const unsigned char* mask = (const unsigned char*)d_in[4]; // jnp.bool_ = 1 byte

  float* out  = (float*)d_out;                               // [B,L,D] first
  float* attn = out + (size_t)BATCH * LSEQ * DHEAD;          // then [B,L,L]

  dim3 grid(BATCH * (LSEQ / QT));   // 2048 workgroups
  sdpa_wmma_f32_kernel<<<grid, 256, 0, stream>>>(q, k, v, diag, mask, out, attn);
}